// Warping_48172353192205
// MI455X (gfx1250) — compile-verified
//
#include <hip/hip_runtime.h>

// dense_image_warp for image [B,256,256,64] f32, flow [B,256,256,2] f32.
// Pure bandwidth-bound gather: 16 lanes per pixel, one float4 (4 channels)
// per lane. Non-temporal hints on streaming data (flow in, output out),
// regular-temporal on image gathers (heavy corner reuse in WGP$/L2).

typedef float v4f __attribute__((ext_vector_type(4)));
typedef float v2f __attribute__((ext_vector_type(2)));

#ifndef WARP_H
#define WARP_H 256
#define WARP_W 256
#define WARP_C 64
#endif

__global__ __launch_bounds__(256) void warp_bilinear_nhwc64(
    const float* __restrict__ image,
    const float* __restrict__ flow,
    float* __restrict__ out,
    int npix)  // B*H*W
{
    const int gid = blockIdx.x * blockDim.x + threadIdx.x;
    const int p   = gid >> 4;       // pixel id
    const int cg  = gid & 15;       // channel group: 4 floats each
    if (p >= npix) return;

    // decode (b, y, x) — H=W=256 so shifts/masks
    const int x = p & (WARP_W - 1);
    const int y = (p >> 8) & (WARP_H - 1);
    const int b = p >> 16;

    // flow(y_comp, x_comp): same 8B address across the 16 lanes of this pixel,
    // collapses to one broadcast request. Streamed once -> non-temporal.
    const v2f f = __builtin_nontemporal_load((const v2f*)flow + p);

    const float qy = (float)y - f.x;
    const float qx = (float)x - f.y;

    float fy = floorf(qy); fy = fminf(fmaxf(fy, 0.0f), (float)(WARP_H - 2));
    float fx = floorf(qx); fx = fminf(fmaxf(fx, 0.0f), (float)(WARP_W - 2));
    const float ay = fminf(fmaxf(qy - fy, 0.0f), 1.0f);
    const float ax = fminf(fmaxf(qx - fx, 0.0f), 1.0f);

    const int iy = (int)fy;
    const int ix = (int)fx;

    // base element index of (b, iy, ix, cg*4)
    const long long base =
        ((long long)((b * WARP_H + iy) * WARP_W + ix)) * WARP_C + (cg << 2);
    const v4f* ip = (const v4f*)(image + base);

    // four corners, each a coalesced b128 load (16 lanes span 256B/corner-row)
    const v4f tl = ip[0];
    const v4f tr = ip[WARP_C / 4];                    // +C floats  (x+1)
    const v4f bl = ip[(WARP_W * WARP_C) / 4];         // +W*C floats (y+1)
    const v4f br = ip[(WARP_W * WARP_C + WARP_C) / 4];

    // top = tl + ax*(tr-tl); bot = bl + ax*(br-bl); out = top + ay*(bot-top)
    v4f top, bot, o;
    top.x = __builtin_fmaf(ax, tr.x - tl.x, tl.x);
    top.y = __builtin_fmaf(ax, tr.y - tl.y, tl.y);
    top.z = __builtin_fmaf(ax, tr.z - tl.z, tl.z);
    top.w = __builtin_fmaf(ax, tr.w - tl.w, tl.w);
    bot.x = __builtin_fmaf(ax, br.x - bl.x, bl.x);
    bot.y = __builtin_fmaf(ax, br.y - bl.y, bl.y);
    bot.z = __builtin_fmaf(ax, br.z - bl.z, bl.z);
    bot.w = __builtin_fmaf(ax, br.w - bl.w, bl.w);
    o.x = __builtin_fmaf(ay, bot.x - top.x, top.x);
    o.y = __builtin_fmaf(ay, bot.y - top.y, top.y);
    o.z = __builtin_fmaf(ay, bot.z - top.z, top.z);
    o.w = __builtin_fmaf(ay, bot.w - top.w, top.w);

    // write-once output -> non-temporal b128 store, fully coalesced
    __builtin_nontemporal_store(o, (v4f*)out + gid);
}

extern "C" void kernel_launch(void* const* d_in, const int* in_sizes, int n_in,
                              void* d_out, int out_size, void* d_ws, size_t ws_size,
                              hipStream_t stream) {
    const float* image = (const float*)d_in[0];
    const float* flow  = (const float*)d_in[1];
    float* out = (float*)d_out;

    const int npix = in_sizes[1] / 2;                 // B*H*W
    const long long nthreads = (long long)npix * 16;  // 16 lanes per pixel
    const int block = 256;                            // 8 wave32 per block
    const int grid  = (int)((nthreads + block - 1) / block);

    warp_bilinear_nhwc64<<<grid, block, 0, stream>>>(image, flow, out, npix);
}